// SelfRegLoss_59356448031037
// MI455X (gfx1250) — compile-verified
//
#include <hip/hip_runtime.h>
#include <math.h>

typedef __attribute__((ext_vector_type(2))) float v2f;
typedef __attribute__((ext_vector_type(8))) float v8f;

#define LAMBDA_SCR 0.015f
#define LAMBDA_IFD 0.015f

// ---------------- workspace layout (in floats) -------------------------
// accum[0] : CE negative-logp sum
// accum[1] : scr0 squared-diff sum     accum[2] : scr1 squared-diff sum
// accum[3..5] : ifd squared-diff sums for feat0..feat2
// norms  = ws + 8        (14336 floats: feat0[0,2048) feat1[2048,6144) feat2[6144,14336))
// flags  = (int*)(ws + 8 + 14336)   same layout, +1 strong / -1 weak
// -----------------------------------------------------------------------

__global__ void init_accum_kernel(float* accum) {
    if (threadIdx.x < 8) accum[threadIdx.x] = 0.0f;
}

// ---- cross entropy: one block per row (32 rows x 1000 logits) ---------
__global__ void ce_kernel(const float* __restrict__ pred,
                          const int* __restrict__ target, float* accum) {
    __shared__ float red[256];
    const int b   = blockIdx.x;
    const int tid = threadIdx.x;
    const float* row = pred + b * 1000;

    float mx = -3.4e38f;
    for (int i = tid; i < 1000; i += 256) mx = fmaxf(mx, row[i]);
    red[tid] = mx; __syncthreads();
    for (int s = 128; s > 0; s >>= 1) {
        if (tid < s) red[tid] = fmaxf(red[tid], red[tid + s]);
        __syncthreads();
    }
    mx = red[0]; __syncthreads();

    float se = 0.0f;
    for (int i = tid; i < 1000; i += 256) se += __expf(row[i] - mx);
    red[tid] = se; __syncthreads();
    for (int s = 128; s > 0; s >>= 1) {
        if (tid < s) red[tid] += red[tid + s];
        __syncthreads();
    }
    if (tid == 0) {
        int t = target[b];
        float logp = row[t] - mx - __logf(red[0]);
        atomicAdd(&accum[0], -logp);
    }
}

// ---- pass 1: per-(b,c) sum of squares (ordering-equivalent to L2 norm)
__global__ void sumsq_kernel(const float* __restrict__ feat, int HW,
                             float* __restrict__ norms_out) {
    __shared__ float red[256];
    const int bc = blockIdx.x;
    const float4* p = (const float4*)(feat + (size_t)bc * HW);
    const int n4 = HW >> 2;
    float s = 0.0f;
    for (int i = threadIdx.x; i < n4; i += 256) {
        if (i + 256 < n4) __builtin_prefetch(&p[i + 256], 0, 1);  // global_prefetch_b8
        float4 v = p[i];
        s += v.x * v.x + v.y * v.y + v.z * v.z + v.w * v.w;
    }
    red[threadIdx.x] = s; __syncthreads();
    for (int st = 128; st > 0; st >>= 1) {
        if (threadIdx.x < st) red[threadIdx.x] += red[threadIdx.x + st];
        __syncthreads();
    }
    if (threadIdx.x == 0) norms_out[bc] = red[0];
}

// ---- stable-argsort ranks (ties -> lower index wins), flag strong/weak
__global__ void flags_kernel(const float* __restrict__ norms, int C,
                             int* __restrict__ flags) {
    __shared__ float ns[256];
    const int b = blockIdx.x;
    const int c = threadIdx.x;
    if (c < C) ns[c] = norms[b * C + c];
    __syncthreads();
    if (c < C) {
        float nc = ns[c];
        int rank = 0;
        for (int j = 0; j < C; ++j) {
            float nj = ns[j];
            rank += (nj > nc) || (nj == nc && j < c);
        }
        int half = C >> 1;
        flags[b * C + c] = (rank < half) ? 1 : ((rank < (half << 1)) ? -1 : 0);
    }
}

// ---- pass 2: IFD strong/weak channel means via V_WMMA_F32_16X16X4_F32.
// Each wave owns 16 consecutive spatial positions of one batch; the K=channels
// GEMM runs against a 2-column selection mask staged in LDS.
// A (16x4 f32): lane L<16 -> (M=L,K=0/1), lane L+16 -> (M=L,K=2/3).
// B (4x16 f32): N=0 column = strong mask, N=1 column = weak mask, rest 0
//               (built branchlessly from loop-invariant lane-role selectors).
// D (16x16 f32): lanes 0/16 hold column N=0 (strong sums), lanes 1/17 N=1.
__global__ void ifd_wmma_kernel(const float* __restrict__ feat,
                                const int* __restrict__ flags,
                                int C, int HW, int groupsPerB,
                                float invHalf, float* slot) {
    __shared__ float2 maskPair[256];   // (strong, weak) per channel
    __shared__ float  wred[8];

    const int lane   = threadIdx.x & 31;
    const int wave   = threadIdx.x >> 5;
    const int group0 = blockIdx.x * 8;          // 8 waves per 256-thread block
    const int b      = group0 / groupsPerB;     // groupsPerB % 8 == 0 -> whole
                                                // block shares one batch
    if (threadIdx.x < C) {
        int f = flags[b * C + threadIdx.x];
        maskPair[threadIdx.x] =
            make_float2(f == 1 ? 1.0f : 0.0f, f == -1 ? 1.0f : 0.0f);
    }
    __syncthreads();

    const int group = group0 + wave;
    const int sp0   = (group - b * groupsPerB) << 4;
    const int m     = lane & 15;          // M (spatial) / N index for B
    const int kb    = (lane >> 4) << 1;   // K base: 0 (lanes 0-15), 2 (16-31)
    const float isS = (m == 0) ? 1.0f : 0.0f;   // loop-invariant lane roles
    const float isW = (m == 1) ? 1.0f : 0.0f;

    const float* fb = feat + (size_t)b * C * HW + sp0 + m;

    v8f acc = {};
    for (int c0 = 0; c0 < C; c0 += 4) {
        v2f A;
        A.x = fb[(c0 + kb)     * HW];
        A.y = fb[(c0 + kb + 1) * HW];
        float2 p0 = maskPair[c0 + kb];
        float2 p1 = maskPair[c0 + kb + 1];
        v2f Bm;
        Bm.x = isS * p0.x + isW * p0.y;   // branchless: 2 v_fma, no EXEC churn
        Bm.y = isS * p1.x + isW * p1.y;
        acc = __builtin_amdgcn_wmma_f32_16x16x4_f32(
            false, A, false, Bm, (short)0, acc, false, false);
    }

    // D[M][0]=strong sum, D[M][1]=weak sum. M=v in lanes 0/1, M=v+8 in 16/17.
    float local = 0.0f;
#pragma unroll
    for (int v = 0; v < 8; ++v) {
        float sLo = __shfl(acc[v], 0),  wLo = __shfl(acc[v], 1);
        float sHi = __shfl(acc[v], 16), wHi = __shfl(acc[v], 17);
        float as = 1.0f / (1.0f + __expf(-sLo * invHalf));
        float aw = 1.0f / (1.0f + __expf(-wLo * invHalf));
        float d  = aw - as;
        local += d * d;
        as = 1.0f / (1.0f + __expf(-sHi * invHalf));
        aw = 1.0f / (1.0f + __expf(-wHi * invHalf));
        d  = aw - as;
        local += d * d;
    }
    if (lane == 0) wred[wave] = local;
    __syncthreads();
    if (threadIdx.x == 0) {
        float s = 0.0f;
#pragma unroll
        for (int i = 0; i < 8; ++i) s += wred[i];
        atomicAdd(slot, s);   // one atomic per block
    }
}

// ---- pass 3: SCR MSE vs repeat-upsampled teacher.
// Teacher channel selection: affine bijection on Z_256 (odd multiplier) as a
// deterministic stand-in for jax.random.permutation (injective, fixed).
__global__ void scr_kernel(const float* __restrict__ student,
                           const float* __restrict__ teacher,
                           int logC, int logHW, int logW, int shift,
                           int permMul, int permAdd, int total4, float* slot) {
    __shared__ float red[256];
    float local = 0.0f;
    const int stride = gridDim.x * blockDim.x;
    for (int u = blockIdx.x * blockDim.x + threadIdx.x; u < total4; u += stride) {
        if (u + stride < total4)
            __builtin_prefetch(student + ((size_t)(u + stride) << 2), 0, 1);
        int e  = u << 2;
        int b  = e >> (logC + logHW);
        int c  = (e >> logHW) & ((1 << logC) - 1);
        int hw = e & ((1 << logHW) - 1);
        int h  = hw >> logW;
        int w  = hw & ((1 << logW) - 1);

        float4 sv = *(const float4*)(student + e);
        int pc = (c * permMul + permAdd) & 255;
        const float* trow = teacher + (((b << 8) + pc) * 32 + (h >> shift)) * 32;
        float t0 = trow[(w)     >> shift];
        float t1 = trow[(w + 1) >> shift];
        float t2 = trow[(w + 2) >> shift];
        float t3 = trow[(w + 3) >> shift];
        float d0 = sv.x - t0, d1 = sv.y - t1, d2 = sv.z - t2, d3 = sv.w - t3;
        local += d0 * d0 + d1 * d1 + d2 * d2 + d3 * d3;
    }
    red[threadIdx.x] = local; __syncthreads();
    for (int s = 128; s > 0; s >>= 1) {
        if (threadIdx.x < s) red[threadIdx.x] += red[threadIdx.x + s];
        __syncthreads();
    }
    if (threadIdx.x == 0) atomicAdd(slot, red[0]);
}

__global__ void finalize_kernel(const float* __restrict__ accum, float* out) {
    if (threadIdx.x == 0 && blockIdx.x == 0) {
        float l_main = accum[0] / 32.0f;
        float scr0 = accum[1] / (float)(32 * 64 * 128 * 128);
        float scr1 = accum[2] / (float)(32 * 128 * 64 * 64);
        float l_scr = 0.5f * (scr0 + scr1);
        float ifd0 = accum[3] / (float)(32 * 128 * 128);
        float ifd1 = accum[4] / (float)(32 * 64 * 64);
        float ifd2 = accum[5] / (float)(32 * 32 * 32);
        float l_ifd = (ifd0 + ifd1 + ifd2) / 3.0f;
        float total = l_main + LAMBDA_SCR * l_scr + LAMBDA_IFD * l_ifd;
        out[0] = total; out[1] = l_main; out[2] = l_scr; out[3] = l_ifd;
    }
}

extern "C" void kernel_launch(void* const* d_in, const int* in_sizes, int n_in,
                              void* d_out, int out_size, void* d_ws, size_t ws_size,
                              hipStream_t stream) {
    (void)in_sizes; (void)n_in; (void)out_size; (void)ws_size;
    const float* pred   = (const float*)d_in[0];
    const float* feat0  = (const float*)d_in[1];   // (32, 64,128,128)
    const float* feat1  = (const float*)d_in[2];   // (32,128, 64, 64)
    const float* feat2  = (const float*)d_in[3];   // (32,256, 32, 32)
    const int*   target = (const int*)d_in[4];
    float* out = (float*)d_out;

    float* ws    = (float*)d_ws;
    float* accum = ws;                      // 8 floats
    float* norms = ws + 8;                  // 14336 floats
    int*   flags = (int*)(ws + 8 + 14336);  // 14336 ints

    init_accum_kernel<<<1, 32, 0, stream>>>(accum);
    ce_kernel<<<32, 256, 0, stream>>>(pred, target, accum);

    // pass 1: per-channel sumsq (one block per (b,c))
    sumsq_kernel<<<32 *  64, 256, 0, stream>>>(feat0, 128 * 128, norms + 0);
    sumsq_kernel<<<32 * 128, 256, 0, stream>>>(feat1,  64 *  64, norms + 2048);
    sumsq_kernel<<<32 * 256, 256, 0, stream>>>(feat2,  32 *  32, norms + 6144);

    flags_kernel<<<32, 256, 0, stream>>>(norms + 0,    64,  flags + 0);
    flags_kernel<<<32, 256, 0, stream>>>(norms + 2048, 128, flags + 2048);
    flags_kernel<<<32, 256, 0, stream>>>(norms + 6144, 256, flags + 6144);

    // pass 2: WMMA selection-GEMM IFD.  groups = B*HW/16, 8 waves/block.
    ifd_wmma_kernel<<<4096, 256, 0, stream>>>(feat0, flags + 0,    64, 128 * 128,
                                              1024, 1.0f /  32.0f, &accum[3]);
    ifd_wmma_kernel<<<1024, 256, 0, stream>>>(feat1, flags + 2048, 128,  64 * 64,
                                              256,  1.0f /  64.0f, &accum[4]);
    ifd_wmma_kernel<<< 256, 256, 0, stream>>>(feat2, flags + 6144, 256,  32 * 32,
                                              64,   1.0f / 128.0f, &accum[5]);

    // pass 3: SCR MSE (student0: 4x repeat, student1: 2x repeat)
    scr_kernel<<<2048, 256, 0, stream>>>(feat0, feat2, 6, 14, 7, 2,
                                         97, 13, 32 * 64 * 128 * 128 / 4, &accum[1]);
    scr_kernel<<<2048, 256, 0, stream>>>(feat1, feat2, 7, 12, 6, 1,
                                         171, 29, 32 * 128 * 64 * 64 / 4, &accum[2]);

    finalize_kernel<<<1, 64, 0, stream>>>(accum, out);
}